// TFNModel_81999515615297
// MI455X (gfx1250) — compile-verified
//
#include <hip/hip_runtime.h>

typedef float v2f __attribute__((ext_vector_type(2)));
typedef float v8f __attribute__((ext_vector_type(8)));

#define MULT 64
#define SQ3f       1.7320508075688772f
#define INV_SQ3f   0.5773502691896258f
#define INV_FAN_Sf 0.08838834764831845f   /* 1/sqrt(128) */
#define INV_FAN_Vf 0.07216878364870323f   /* 1/sqrt(192) */
#define INV_SQ2f   0.7071067811865476f

__device__ __forceinline__ v8f wmma4(v2f a, v2f b, v8f c) {
  // D = A(16x4,f32) * B(4x16,f32) + C(16x16,f32)
  return __builtin_amdgcn_wmma_f32_16x16x4_f32(false, a, false, b, (short)0, c,
                                               false, false);
}

__device__ __forceinline__ float sigmoidf_(float x) {
  return 1.0f / (1.0f + __expf(-x));
}

// ---------------------------------------------------------------------------
// Repack a row-major [64][ld] weight block into WMMA B-fragment order:
//   packed[((nt*16 + kc)*64) + 2*L + j] = W[(4*kc + 2*(L>>4) + j)*ld + nt*16 + (L&15)]
// so each wave's B fragment for (nt,kc) is one contiguous 256B chunk and each
// lane's two elements are a single b64 load.
// ---------------------------------------------------------------------------
__global__ __launch_bounds__(256) void tfn_pack_kernel(
    const float* __restrict__ W, float* __restrict__ out, int ld)
{
  int p = blockIdx.x * blockDim.x + threadIdx.x;
  int total = 64 * ld;
  if (p >= total) return;
  int nt  = p / (16 * 64);
  int rem = p - nt * (16 * 64);
  int kc  = rem >> 6;
  int q   = rem & 63;
  int L   = q >> 1, j = q & 1;
  int hi  = L >> 4, m = L & 15;
  int k   = 4 * kc + 2 * hi + j;
  out[p] = W[k * ld + nt * 16 + m];
}

// ---------------------------------------------------------------------------
// Edge preprocessing: radial MLP -> g0[e], y1g[e][3]
// ---------------------------------------------------------------------------
__global__ __launch_bounds__(256) void tfn_edge_kernel(
    const float* __restrict__ pos,
    const int* __restrict__ src, const int* __restrict__ dst,
    const float* __restrict__ w1, const float* __restrict__ b1,
    const float* __restrict__ w2, const float* __restrict__ b2,
    const float* __restrict__ w3, const float* __restrict__ b3,
    float* __restrict__ g0v, float* __restrict__ y1g, int E)
{
  __shared__ float w2s[64 * 64];
  for (int i = threadIdx.x; i < 64 * 64; i += blockDim.x) w2s[i] = w2[i];
  __syncthreads();

  int e = blockIdx.x * blockDim.x + threadIdx.x;
  if (e >= E) return;
  int si = src[e], di = dst[e];
  float vx = pos[di * 3 + 0] - pos[si * 3 + 0];
  float vy = pos[di * 3 + 1] - pos[si * 3 + 1];
  float vz = pos[di * 3 + 2] - pos[si * 3 + 2];
  float r = sqrtf(vx * vx + vy * vy + vz * vz);
  float inv = 1.0f / (r + 1e-9f);

  float h1[64];
  #pragma unroll
  for (int j = 0; j < 64; ++j) h1[j] = fmaxf(r * w1[j] + b1[j], 0.0f);

  float gA = b3[0], gB = b3[1];
  for (int j = 0; j < 64; ++j) {
    float a = b2[j];
    #pragma unroll 8
    for (int i = 0; i < 64; ++i) a += h1[i] * w2s[i * 64 + j];
    a = fmaxf(a, 0.0f);
    gA += a * w3[j * 2 + 0];
    gB += a * w3[j * 2 + 1];
  }
  g0v[e] = gA;
  float sc = SQ3f * inv * gB;
  y1g[e * 3 + 0] = vx * sc;
  y1g[e * 3 + 1] = vy * sc;
  y1g[e * 3 + 2] = vz * sc;
}

// ---------------------------------------------------------------------------
// Init s = emb[atoms]
// ---------------------------------------------------------------------------
__global__ void tfn_init_s_kernel(const int* __restrict__ atoms,
                                  const float* __restrict__ emb,
                                  float* __restrict__ s0, int N)
{
  int i = blockIdx.x * blockDim.x + threadIdx.x;
  if (i < N * MULT) {
    int n = i >> 6, u = i & 63;
    s0[i] = emb[atoms[n] * MULT + u];
  }
}

__global__ void tfn_zero_kernel(float* __restrict__ p, size_t n)
{
  size_t i = (size_t)blockIdx.x * blockDim.x + threadIdx.x;
  size_t stride = (size_t)gridDim.x * blockDim.x;
  for (; i < n; i += stride) p[i] = 0.0f;
}

// ---------------------------------------------------------------------------
// Message kernel: 1 wave (32 lanes) handles a 16-edge tile.
// WMMA f32 16x16x4 fragments:
//   A: lane L -> row m=L&15, elems K = 4*kc + 2*(L>>4) + {0,1}
//   B: single b64 from packed weights (see tfn_pack_kernel)
//   C: VGPR r, lane L -> (M = r + 8*(L>>4), N = L&15)
// ---------------------------------------------------------------------------
__global__ __launch_bounds__(32) void tfn_message_kernel(
    const float* __restrict__ s_cur,  // [N][64]
    const float* __restrict__ v_cur,  // [3][N][64]
    const int* __restrict__ src, const int* __restrict__ dst,
    const float* __restrict__ g0v, const float* __restrict__ y1g,
    const float* __restrict__ Ws0p,   // packed [8][16][64]  (64x128)
    const float* __restrict__ Ws1p,   // packed [8][16][64]
    const float* __restrict__ Wv0p,   // packed [4][16][64]  (64x64)
    const float* __restrict__ Wv1p,   // packed [4][16][64]
    const float* __restrict__ Wv2p,   // packed [4][16][64]
    float* __restrict__ Sacc,         // [N][128]
    float* __restrict__ Vacc,         // [3][N][64]
    int N)
{
  const int lane = threadIdx.x;
  const int m   = lane & 15;
  const int hi  = lane >> 4;
  const int ka  = hi * 2;
  const int bofs = lane * 2;          // packed B-fragment lane offset
  const int e0  = blockIdx.x * 16;

  const int   eA  = e0 + m;
  const int   sN  = src[eA];
  const float g0m = g0v[eA];
  const float yg0 = y1g[eA * 3 + 0];
  const float yg1 = y1g[eA * 3 + 1];
  const float yg2 = y1g[eA * 3 + 2];

  const float* sRow  = s_cur + (size_t)sN * MULT;
  const float* vRow0 = v_cur + (size_t)sN * MULT;
  const float* vRow1 = v_cur + ((size_t)N + sN) * MULT;
  const float* vRow2 = v_cur + ((size_t)2 * N + sN) * MULT;

  // Epilogue row data (rows M = r + 8*hi of this tile)
  int   dR[8];
  float yR[8][3];
  #pragma unroll
  for (int r = 0; r < 8; ++r) {
    int er = e0 + r + 8 * hi;
    dR[r] = dst[er];
    yR[r][0] = y1g[er * 3 + 0];
    yR[r][1] = y1g[er * 3 + 1];
    yR[r][2] = y1g[er * 3 + 2];
  }

  // ---- t0 = s_e @ Wv0  (16x64) : kept in registers ----
  v8f t0[4];
  #pragma unroll
  for (int nt = 0; nt < 4; ++nt) {
    v8f acc = {};
    for (int kc = 0; kc < 16; ++kc) {
      int kb = 4 * kc + ka;
      v2f a = *(const v2f*)(sRow + kb);
      v2f b = *(const v2f*)(Wv0p + (nt * 16 + kc) * 64 + bofs);
      acc = wmma4(a, b, acc);
    }
    t0[nt] = acc;
  }

  // ---- vector messages, per cartesian channel c ----
  #pragma unroll
  for (int c = 0; c < 3; ++c) {
    const int c1 = (c + 1) % 3, c2 = (c + 2) % 3;
    const float* vrC  = (c  == 0) ? vRow0 : ((c  == 1) ? vRow1 : vRow2);
    const float* vrC1 = (c1 == 0) ? vRow0 : ((c1 == 1) ? vRow1 : vRow2);
    const float* vrC2 = (c2 == 0) ? vRow0 : ((c2 == 1) ? vRow1 : vRow2);
    const float ygc1 = (c1 == 0) ? yg0 : ((c1 == 1) ? yg1 : yg2);
    const float ygc2 = (c2 == 0) ? yg0 : ((c2 == 1) ? yg1 : yg2);
    float* Vc = Vacc + (size_t)c * N * MULT;

    #pragma unroll
    for (int nt = 0; nt < 4; ++nt) {
      v8f acc = {};
      for (int kc = 0; kc < 16; ++kc) {
        int kb = 4 * kc + ka;
        // (v_c * g0) @ Wv1
        v2f avg = *(const v2f*)(vrC + kb);
        avg.x *= g0m; avg.y *= g0m;
        v2f bw1 = *(const v2f*)(Wv1p + (nt * 16 + kc) * 64 + bofs);
        acc = wmma4(avg, bw1, acc);
        // (cross(v, y1g)_c / sqrt2) @ Wv2
        v2f aa = *(const v2f*)(vrC1 + kb);
        v2f bb = *(const v2f*)(vrC2 + kb);
        v2f ax;
        ax.x = (aa.x * ygc2 - bb.x * ygc1) * INV_SQ2f;
        ax.y = (aa.y * ygc2 - bb.y * ygc1) * INV_SQ2f;
        v2f bw2 = *(const v2f*)(Wv2p + (nt * 16 + kc) * 64 + bofs);
        acc = wmma4(ax, bw2, acc);
      }
      const int col = nt * 16 + m;
      #pragma unroll
      for (int r = 0; r < 8; ++r) {
        float val = (t0[nt][r] * yR[r][c] + acc[r]) * INV_FAN_Vf;
        atomicAdd(&Vc[(size_t)dR[r] * MULT + col], val);
      }
    }
  }

  // ---- scalar messages: (s_e*g0)@Ws0 + (dot/sqrt3)@Ws1, 16x128 ----
  #pragma unroll
  for (int nt = 0; nt < 8; ++nt) {
    v8f acc = {};
    for (int kc = 0; kc < 16; ++kc) {
      int kb = 4 * kc + ka;
      v2f as = *(const v2f*)(sRow + kb);
      as.x *= g0m; as.y *= g0m;
      v2f b0 = *(const v2f*)(Ws0p + (nt * 16 + kc) * 64 + bofs);
      acc = wmma4(as, b0, acc);

      v2f a0 = *(const v2f*)(vRow0 + kb);
      v2f a1 = *(const v2f*)(vRow1 + kb);
      v2f a2 = *(const v2f*)(vRow2 + kb);
      v2f ad;
      ad.x = (a0.x * yg0 + a1.x * yg1 + a2.x * yg2) * INV_SQ3f;
      ad.y = (a0.y * yg0 + a1.y * yg1 + a2.y * yg2) * INV_SQ3f;
      v2f bw = *(const v2f*)(Ws1p + (nt * 16 + kc) * 64 + bofs);
      acc = wmma4(ad, bw, acc);
    }
    const int col = nt * 16 + m;
    #pragma unroll
    for (int r = 0; r < 8; ++r)
      atomicAdd(&Sacc[(size_t)dR[r] * 128 + col], acc[r] * INV_FAN_Sf);
  }
}

// ---------------------------------------------------------------------------
// Node update: self-interaction WMMA + SiLU / sigmoid gating.
// ---------------------------------------------------------------------------
__global__ __launch_bounds__(32) void tfn_node_update_kernel(
    const float* __restrict__ s_cur, const float* __restrict__ v_cur,
    const float* __restrict__ Sacc, const float* __restrict__ Vacc,
    const float* __restrict__ Wssp,  // packed [8][16][64] (64x128)
    const float* __restrict__ Wsvp,  // packed [4][16][64] (64x64)
    float* __restrict__ s_new, float* __restrict__ v_new, int N)
{
  const int lane = threadIdx.x;
  const int m = lane & 15, hi = lane >> 4;
  const int ka = hi * 2;
  const int bofs = lane * 2;
  const int n0 = blockIdx.x * 16;
  const int nodeA = n0 + m;
  const float* sRow = s_cur + (size_t)nodeA * MULT;

  v8f gate[4];

  #pragma unroll
  for (int nt = 0; nt < 8; ++nt) {
    v8f acc = {};
    for (int kc = 0; kc < 16; ++kc) {
      int kb = 4 * kc + ka;
      v2f a = *(const v2f*)(sRow + kb);
      v2f b = *(const v2f*)(Wssp + (nt * 16 + kc) * 64 + bofs);
      acc = wmma4(a, b, acc);
    }
    const int col = nt * 16 + m;
    #pragma unroll
    for (int r = 0; r < 8; ++r) {
      int node = n0 + r + 8 * hi;
      float tot = acc[r] + Sacc[(size_t)node * 128 + col];
      if (nt < 4) {
        s_new[(size_t)node * MULT + col] = tot * sigmoidf_(tot);  // SiLU
      } else {
        gate[nt - 4][r] = sigmoidf_(tot);
      }
    }
  }

  #pragma unroll
  for (int c = 0; c < 3; ++c) {
    const float* vRow = v_cur + ((size_t)c * N + nodeA) * MULT;
    const float* Vc = Vacc + (size_t)c * N * MULT;
    float* vNc = v_new + (size_t)c * N * MULT;
    #pragma unroll
    for (int nt = 0; nt < 4; ++nt) {
      v8f acc = {};
      for (int kc = 0; kc < 16; ++kc) {
        int kb = 4 * kc + ka;
        v2f a = *(const v2f*)(vRow + kb);
        v2f b = *(const v2f*)(Wsvp + (nt * 16 + kc) * 64 + bofs);
        acc = wmma4(a, b, acc);
      }
      const int col = nt * 16 + m;
      #pragma unroll
      for (int r = 0; r < 8; ++r) {
        int node = n0 + r + 8 * hi;
        float tot = acc[r] + Vc[(size_t)node * MULT + col];
        vNc[(size_t)node * MULT + col] = gate[nt][r] * tot;
      }
    }
  }
}

// ---------------------------------------------------------------------------
// Mean-pool over nodes (64 columns), then tiny head MLP.
// ---------------------------------------------------------------------------
__global__ __launch_bounds__(256) void tfn_pool_kernel(
    const float* __restrict__ s, float* __restrict__ pooled, int N)
{
  int gid = blockIdx.x * blockDim.x + threadIdx.x;  // 64*256 threads
  int u = gid & 63;
  int part = gid >> 6;  // 0..255
  float sum = 0.0f;
  for (int n = part; n < N; n += 256) sum += s[(size_t)n * MULT + u];
  atomicAdd(&pooled[u], sum);
}

__global__ __launch_bounds__(64) void tfn_head_kernel(
    const float* __restrict__ pooled,
    const float* __restrict__ w1, const float* __restrict__ b1,
    const float* __restrict__ w2, const float* __restrict__ b2,
    float* __restrict__ out, int N)
{
  __shared__ float red[64];
  int j = threadIdx.x;
  float invN = 1.0f / (float)N;
  float a = b1[j];
  #pragma unroll 8
  for (int i = 0; i < 64; ++i) a += (pooled[i] * invN) * w1[i * 64 + j];
  a = fmaxf(a, 0.0f);
  red[j] = a * w2[j];
  __syncthreads();
  #pragma unroll
  for (int off = 32; off > 0; off >>= 1) {
    if (j < off) red[j] += red[j + off];
    __syncthreads();
  }
  if (j == 0) out[0] = red[0] + b2[0];
}

// ---------------------------------------------------------------------------
// Host launcher
// ---------------------------------------------------------------------------
extern "C" void kernel_launch(void* const* d_in, const int* in_sizes, int n_in,
                              void* d_out, int out_size, void* d_ws, size_t ws_size,
                              hipStream_t stream)
{
  (void)n_in; (void)out_size; (void)ws_size;
  const int*   atoms   = (const int*)d_in[0];
  const float* pos     = (const float*)d_in[1];
  const int*   eidx    = (const int*)d_in[2];
  const float* emb     = (const float*)d_in[3];
  const float* rmlp_w1 = (const float*)d_in[4];
  const float* rmlp_b1 = (const float*)d_in[5];
  const float* rmlp_w2 = (const float*)d_in[6];
  const float* rmlp_b2 = (const float*)d_in[7];
  const float* rmlp_w3 = (const float*)d_in[8];
  const float* rmlp_b3 = (const float*)d_in[9];
  const float* tp_w_s  = (const float*)d_in[10];
  const float* tp_w_v  = (const float*)d_in[11];
  const float* si_w_s  = (const float*)d_in[12];
  const float* si_w_v  = (const float*)d_in[13];
  const float* pred_w1 = (const float*)d_in[14];
  const float* pred_b1 = (const float*)d_in[15];
  const float* pred_w2 = (const float*)d_in[16];
  const float* pred_b2 = (const float*)d_in[17];

  const int N = in_sizes[0];
  const int E = in_sizes[2] / 2;
  const int* src = eidx;
  const int* dst = eidx + E;

  float* ws = (float*)d_ws;
  size_t off = 0;
  auto alloc = [&](size_t nfloats) -> float* {
    float* p = ws + off;
    off += (nfloats + 63) & ~(size_t)63;  // 256B-aligned chunks
    return p;
  };
  float* g0v    = alloc((size_t)E);
  float* y1g    = alloc((size_t)3 * E);
  float* sA     = alloc((size_t)N * MULT);
  float* sB     = alloc((size_t)N * MULT);
  float* vA     = alloc((size_t)3 * N * MULT);
  float* vB     = alloc((size_t)3 * N * MULT);
  float* Sacc   = alloc((size_t)N * 128);
  float* Vacc   = alloc((size_t)3 * N * MULT);
  float* pooled = alloc(64);
  float* pWs    = alloc(2 * 2 * 8192);  // packed tp_w_s  (per layer: Ws0,Ws1)
  float* pWv    = alloc(2 * 3 * 4096);  // packed tp_w_v  (per layer: Wv0..2)
  float* pWss   = alloc(2 * 8192);      // packed si_w_s
  float* pWsv   = alloc(2 * 4096);      // packed si_w_v

  // 0) repack weights into WMMA B-fragment order (coalesced b64 loads)
  for (int l = 0; l < 2; ++l) {
    for (int i = 0; i < 2; ++i)
      tfn_pack_kernel<<<(64 * 128 + 255) / 256, 256, 0, stream>>>(
          tp_w_s + ((size_t)l * 2 + i) * 8192, pWs + ((size_t)l * 2 + i) * 8192, 128);
    for (int i = 0; i < 3; ++i)
      tfn_pack_kernel<<<(64 * 64 + 255) / 256, 256, 0, stream>>>(
          tp_w_v + ((size_t)l * 3 + i) * 4096, pWv + ((size_t)l * 3 + i) * 4096, 64);
    tfn_pack_kernel<<<(64 * 128 + 255) / 256, 256, 0, stream>>>(
        si_w_s + (size_t)l * 8192, pWss + (size_t)l * 8192, 128);
    tfn_pack_kernel<<<(64 * 64 + 255) / 256, 256, 0, stream>>>(
        si_w_v + (size_t)l * 4096, pWsv + (size_t)l * 4096, 64);
  }

  // 1) edge features
  tfn_edge_kernel<<<(E + 255) / 256, 256, 0, stream>>>(
      pos, src, dst, rmlp_w1, rmlp_b1, rmlp_w2, rmlp_b2, rmlp_w3, rmlp_b3,
      g0v, y1g, E);

  // 2) init node state
  tfn_init_s_kernel<<<(N * MULT + 255) / 256, 256, 0, stream>>>(atoms, emb, sA, N);
  tfn_zero_kernel<<<1024, 256, 0, stream>>>(vA, (size_t)3 * N * MULT);

  float* s_cur = sA; float* v_cur = vA;
  float* s_nxt = sB; float* v_nxt = vB;

  for (int l = 0; l < 2; ++l) {
    tfn_zero_kernel<<<1024, 256, 0, stream>>>(Sacc, (size_t)N * 128);
    tfn_zero_kernel<<<1024, 256, 0, stream>>>(Vacc, (size_t)3 * N * MULT);

    tfn_message_kernel<<<E / 16, 32, 0, stream>>>(
        s_cur, v_cur, src, dst, g0v, y1g,
        pWs + (size_t)l * 2 * 8192, pWs + (size_t)l * 2 * 8192 + 8192,
        pWv + (size_t)l * 3 * 4096, pWv + (size_t)l * 3 * 4096 + 4096,
        pWv + (size_t)l * 3 * 4096 + 8192,
        Sacc, Vacc, N);

    tfn_node_update_kernel<<<N / 16, 32, 0, stream>>>(
        s_cur, v_cur, Sacc, Vacc,
        pWss + (size_t)l * 8192, pWsv + (size_t)l * 4096,
        s_nxt, v_nxt, N);

    float* ts = s_cur; s_cur = s_nxt; s_nxt = ts;
    float* tv = v_cur; v_cur = v_nxt; v_nxt = tv;
  }

  // 3) mean pool + head
  tfn_zero_kernel<<<1, 64, 0, stream>>>(pooled, 64);
  tfn_pool_kernel<<<64, 256, 0, stream>>>(s_cur, pooled, N);
  tfn_head_kernel<<<1, 64, 0, stream>>>(pooled, pred_w1, pred_b1, pred_w2, pred_b2,
                                        (float*)d_out, N);
}